// SimpleSpline_44598940401671
// MI455X (gfx1250) — compile-verified
//
#include <hip/hip_runtime.h>

// Cubic B-spline (uniform knots) elementwise evaluation, 8192x8192 fp32.
// Memory-bound: 512 MB HBM traffic -> ~22 us floor at 23.3 TB/s on MI455X.
// Strategy:
//   * per-block precompute of 29 x 4 monomial coeffs in LDS (16B rows)
//   * one ds_load_b128 gather + 3-FMA Horner per element
//   * 128-bit non-temporal global loads/stores (stream past 192MB L2)
//   * branchless linear extrapolation outside [k0, kl]
//   * 8192 blocks x 256 thr, grid-stride (8 float4/thread) to amortize
//     per-block setup (uniform s_loads, v_div sequences, table build)

#define NUM_KNOTS 30
#define NUM_COEFFS 32          // NUM_KNOTS + 3 - 1
#define NUM_INTERVALS 29       // idx in [0, NUM_KNOTS-2]

typedef float v4f __attribute__((ext_vector_type(4)));

__global__ __launch_bounds__(256) void SimpleSpline_kernel(
    const float* __restrict__ x,
    const float* __restrict__ coeffs,
    const float* __restrict__ knots,
    float* __restrict__ out,
    int n4,          // number of float4 vectors
    int n)           // total elements (for scalar tail)
{
    // 29 rows x 4 floats, 16-byte aligned rows -> ds_load_b128 per lookup
    __shared__ __align__(16) float tab[NUM_INTERVALS * 4];

    // Uniform scalars: compiler emits s_load for these (pointers + offsets uniform)
    const float k0 = knots[0];
    const float kl = knots[NUM_KNOTS - 1];
    const float spacing = (kl - k0) * (1.0f / (NUM_KNOTS - 1));
    const float inv_sp  = 1.0f / spacing;
    const float ls = (coeffs[2] - coeffs[0]) / (2.0f * spacing);
    const float rs = (coeffs[NUM_COEFFS - 1] - coeffs[NUM_COEFFS - 3]) / (2.0f * spacing);

    const int t = threadIdx.x;

    // Build per-interval monomial coefficients:
    // interp(u) = a0 + a1*u + a2*u^2 + a3*u^3
    if (t < NUM_INTERVALS) {
        const float p0 = coeffs[t + 0];
        const float p1 = coeffs[t + 1];
        const float p2 = coeffs[t + 2];
        const float p3 = coeffs[t + 3];
        tab[t * 4 + 0] = (p0 + 4.0f * p1 + p2) * (1.0f / 6.0f);
        tab[t * 4 + 1] = (p2 - p0) * 0.5f;
        tab[t * 4 + 2] = (p0 - 2.0f * p1 + p2) * 0.5f;
        tab[t * 4 + 3] = (p3 - p0 + 3.0f * (p1 - p2)) * (1.0f / 6.0f);
    }
    __syncthreads();

    const v4f* __restrict__ x4 = (const v4f*)x;
    v4f* __restrict__ o4 = (v4f*)out;

    const int stride = (int)(gridDim.x * blockDim.x);
    int i = (int)(blockIdx.x * blockDim.x) + t;

    for (; i < n4; i += stride) {
        v4f v = __builtin_nontemporal_load(&x4[i]);   // global_load_b128 NT
        v4f r;
#pragma unroll
        for (int c = 0; c < 4; ++c) {
            const float xv = v[c];
            // single-op clamp to [k0, kl]
            const float xc = __builtin_amdgcn_fmed3f(xv, k0, kl);
            // grid >= 0 guaranteed (xc >= k0, explicit sub before mul)
            const float grid = (xc - k0) * inv_sp;
            // floor >= 0 already; only upper clamp needed
            const float fl = fminf(__builtin_floorf(grid),
                                   (float)(NUM_INTERVALS - 1));
            const float u = grid - fl;
            const int idx = (int)fl;

            // single ds_load_b128: 16B-aligned row gather
            const v4f a = *(const v4f*)(&tab[idx * 4]);

            // Horner: 3 FMA
            float p = __builtin_fmaf(
                          __builtin_fmaf(__builtin_fmaf(a[3], u, a[2]), u, a[1]),
                          u, a[0]);

            // branchless extrapolation: only one term can be nonzero
            p = __builtin_fmaf(ls, fminf(xv - k0, 0.0f), p);
            p = __builtin_fmaf(rs, fmaxf(xv - kl, 0.0f), p);
            r[c] = p;
        }
        __builtin_nontemporal_store(r, &o4[i]);       // global_store_b128 NT
    }

    // Scalar tail (n not divisible by 4) — inactive for 8192x8192 but safe.
    const int tail_base = n4 * 4;
    const int tid = (int)(blockIdx.x * blockDim.x) + t;
    const int rem = n - tail_base;
    if (tid < rem) {
        const float xv = x[tail_base + tid];
        const float xc = __builtin_amdgcn_fmed3f(xv, k0, kl);
        const float grid = (xc - k0) * inv_sp;
        const float fl = fminf(__builtin_floorf(grid),
                               (float)(NUM_INTERVALS - 1));
        const float u = grid - fl;
        const int idx = (int)fl;
        const v4f a = *(const v4f*)(&tab[idx * 4]);
        float p = __builtin_fmaf(
                      __builtin_fmaf(__builtin_fmaf(a[3], u, a[2]), u, a[1]),
                      u, a[0]);
        p = __builtin_fmaf(ls, fminf(xv - k0, 0.0f), p);
        p = __builtin_fmaf(rs, fmaxf(xv - kl, 0.0f), p);
        out[tail_base + tid] = p;
    }
}

extern "C" void kernel_launch(void* const* d_in, const int* in_sizes, int n_in,
                              void* d_out, int out_size, void* d_ws, size_t ws_size,
                              hipStream_t stream) {
    const float* x      = (const float*)d_in[0];   // (8192*8192,) fp32
    const float* coeffs = (const float*)d_in[1];   // (32,) fp32
    const float* knots  = (const float*)d_in[2];   // (30,) fp32
    float* out = (float*)d_out;

    const int n  = in_sizes[0];
    const int n4 = n / 4;

    const int block = 256;
    // Cap blocks so each thread iterates ~8x -> per-block setup (s_loads,
    // v_div sequences, LDS table build, barrier) amortizes to ~5%.
    long long need = ((long long)n4 + block - 1) / block;
    long long cap  = 8192;
    long long g    = need < cap ? need : cap;
    if (g < 1) g = 1;
    int grid = (int)g;

    SimpleSpline_kernel<<<grid, block, 0, stream>>>(x, coeffs, knots, out, n4, n);
}